// GCNLinkPredictor_32701880991818
// MI455X (gfx1250) — compile-verified
//
#include <hip/hip_runtime.h>
#include <hip/hip_bf16.h>

#define N_IN   256
#define N_HID  128
#define N_LOC  512
#define EPS_BN 1e-5f
#define EPS_COS 1e-8f
#define NEG_SLOPE 0.01f

typedef __attribute__((ext_vector_type(2))) float v2f;
typedef __attribute__((ext_vector_type(8))) float v8f;

// ---------------------------------------------------------------------------
// Small prep kernels
// ---------------------------------------------------------------------------

// Fold BatchNorm (eval) into per-channel affine: xb = x*a + c
__global__ void bn_prep_kernel(const float* __restrict__ gamma,
                               const float* __restrict__ beta,
                               const float* __restrict__ mean,
                               const float* __restrict__ var,
                               float* __restrict__ a, float* __restrict__ c, int n) {
    int i = blockIdx.x * blockDim.x + threadIdx.x;
    if (i < n) {
        float s = gamma[i] * rsqrtf(var[i] + EPS_BN);
        a[i] = s;
        c[i] = beta[i] - mean[i] * s;
    }
}

__global__ void fill_kernel(float* __restrict__ p, float val, int n) {
    int i = blockIdx.x * blockDim.x + threadIdx.x;
    if (i < n) p[i] = val;
}

// deg_s[dst]++ ; deg_d[src]++  (self loops pre-seeded via fill(1.0))
__global__ void degree_kernel(const int* __restrict__ src, const int* __restrict__ dst,
                              float* __restrict__ deg_s, float* __restrict__ deg_d, int E) {
    int e = blockIdx.x * blockDim.x + threadIdx.x;
    if (e < E) {
        atomicAdd(deg_s + dst[e], 1.0f);
        atomicAdd(deg_d + src[e], 1.0f);
    }
}

__global__ void rsqrt_kernel(float* __restrict__ p, int n) {
    int i = blockIdx.x * blockDim.x + threadIdx.x;
    if (i < n) p[i] = rsqrtf(p[i]);   // deg >= 1 always (self loops)
}

// ---------------------------------------------------------------------------
// WMMA f32 GEMM: C[M,NN] = A[M,K] @ B[K,NN] + bias, optional BN fused on A.
// One 16x16 output tile per wave32, V_WMMA_F32_16X16X4_F32, K stepped by 4.
// A lane layout (ISA 7.12.2, 32-bit A 16x4): lanes 0-15 -> K=0,1 ; lanes 16-31 -> K=2,3
// B lane layout (row striped across lanes): v0: K=0 (lo) / K=2 (hi); v1: K=1 / K=3
// C lane layout: VGPR r: M=r (lanes 0-15), M=r+8 (lanes 16-31), N = lane%16
// ---------------------------------------------------------------------------
template<int K, int NN, bool FUSE_BN>
__global__ void gemm_wmma_f32(const float* __restrict__ A, const float* __restrict__ B,
                              const float* __restrict__ bias,
                              const float* __restrict__ bn_a, const float* __restrict__ bn_c,
                              float* __restrict__ C, int M) {
    const int lane  = threadIdx.x & 31;
    const int wib   = threadIdx.x >> 5;
    const int wpb   = blockDim.x >> 5;
    const int tile  = blockIdx.x * wpb + wib;
    const int ntile = NN / 16;
    const int mtile = (M + 15) >> 4;
    const int mt = tile / ntile;
    const int nt = tile % ntile;
    if (mt >= mtile) return;

    const int rlane = lane & 15;          // row (A) / col (B,C) within tile
    const int khalf = (lane >> 4) << 1;   // 0 or 2
    int arow = mt * 16 + rlane;
    if (arow >= M) arow = M - 1;          // clamp (EXEC must stay all-ones for WMMA)
    const float* ap = A + (size_t)arow * K;
    const float* bp = B + nt * 16 + rlane;

    v8f acc = {};
    for (int k = 0; k < K; k += 4) {
        // Unconditional speculative prefetch of the streamed A row ahead
        // (global_prefetch_b8; overshoot is silently dropped, B stays L2-hot).
        __builtin_prefetch(ap + k + 32, 0, 1);
        float a0 = ap[k + khalf];
        float a1 = ap[k + khalf + 1];
        if (FUSE_BN) {
            a0 = a0 * bn_a[k + khalf]     + bn_c[k + khalf];
            a1 = a1 * bn_a[k + khalf + 1] + bn_c[k + khalf + 1];
        }
        v2f av; av.x = a0; av.y = a1;
        v2f bv;
        bv.x = bp[(size_t)(k + khalf)     * NN];
        bv.y = bp[(size_t)(k + khalf + 1) * NN];
        acc = __builtin_amdgcn_wmma_f32_16x16x4_f32(
            /*neg_a=*/false, av, /*neg_b=*/false, bv,
            /*c_mod=*/(short)0, acc, /*reuse_a=*/false, /*reuse_b=*/false);
    }

    const int col   = nt * 16 + rlane;
    const int rbase = mt * 16 + ((lane >> 4) << 3);
    const float bb  = bias ? bias[col] : 0.0f;
    float* cp = C + (size_t)rbase * NN + col;
    if (rbase + 8 <= M) {
        // Full tile: straight-line stores (single branch for whole epilogue).
#pragma unroll
        for (int r = 0; r < 8; ++r) cp[(size_t)r * NN] = acc[r] + bb;
    } else {
#pragma unroll
        for (int r = 0; r < 8; ++r)
            if (rbase + r < M) cp[(size_t)r * NN] = acc[r] + bb;
    }
}

// ---------------------------------------------------------------------------
// GCN edge scatter: acc[to] += hw[from] * dinv[from] * dinv[to]   (one wave/edge)
// ---------------------------------------------------------------------------
__global__ void scatter_kernel(const float* __restrict__ hw, float* __restrict__ acc,
                               const float* __restrict__ dinv,
                               const int* __restrict__ from, const int* __restrict__ to, int E) {
    const int wave = (blockIdx.x * blockDim.x + threadIdx.x) >> 5;
    const int lane = threadIdx.x & 31;
    if (wave >= E) return;
    const int u = from[wave];
    const int v = to[wave];
    const float coef = dinv[u] * dinv[v];
    const float4 m = ((const float4*)(hw + (size_t)u * N_HID))[lane];
    float* ap = acc + (size_t)v * N_HID + lane * 4;
    atomicAdd(ap + 0, m.x * coef);
    atomicAdd(ap + 1, m.y * coef);
    atomicAdd(ap + 2, m.z * coef);
    atomicAdd(ap + 3, m.w * coef);
}

// out = leaky_relu(acc + hw*dinv^2 (self loop) + bias)
__global__ void finalize_kernel(const float* __restrict__ acc, const float* __restrict__ hw,
                                const float* __restrict__ dinv, const float* __restrict__ bias,
                                float* __restrict__ out, int total) {
    int i = blockIdx.x * blockDim.x + threadIdx.x;
    if (i >= total) return;
    const int node = i >> 7;          // /128
    const int ch   = i & (N_HID - 1);
    const float di = dinv[node];
    float v = acc[i] + hw[i] * di * di + bias[ch];
    out[i] = v >= 0.0f ? v : NEG_SLOPE * v;
}

// ---------------------------------------------------------------------------
// Final per-edge kernel: logits / cosine-similarity (one wave32 per edge)
// ---------------------------------------------------------------------------
__global__ void edge_kernel(const float* __restrict__ s2, const float* __restrict__ r2,
                            const float* __restrict__ loc,
                            const int* __restrict__ src, const int* __restrict__ dst,
                            float* __restrict__ out, int E) {
    const int wave = (blockIdx.x * blockDim.x + threadIdx.x) >> 5;
    const int lane = threadIdx.x & 31;
    if (wave >= E) return;
    const int u = src[wave];
    const int v = dst[wave];

    const float* a = s2 + (size_t)u * N_HID;
    const float* b = r2 + (size_t)v * N_HID;
    float logit = 0.0f;
#pragma unroll
    for (int c = lane; c < N_HID; c += 32) logit += a[c] * b[c];

    const float* la = loc + (size_t)u * N_LOC;
    const float* lb = loc + (size_t)v * N_LOC;
    float dl = 0.0f, na = 0.0f, nb = 0.0f;
#pragma unroll 4
    for (int c = lane; c < N_LOC; c += 32) {
        float x = la[c], y = lb[c];
        dl += x * y; na += x * x; nb += y * y;
    }
#pragma unroll
    for (int off = 16; off > 0; off >>= 1) {
        logit += __shfl_xor(logit, off, 32);
        dl    += __shfl_xor(dl, off, 32);
        na    += __shfl_xor(na, off, 32);
        nb    += __shfl_xor(nb, off, 32);
    }
    if (lane == 0) {
        float denom = fmaxf(sqrtf(na) * sqrtf(nb), EPS_COS);
        float cosv  = dl / denom;
        out[wave] = logit / cosv;
    }
}

// ---------------------------------------------------------------------------
// Host-side orchestration
// ---------------------------------------------------------------------------
static inline int cdiv(int a, int b) { return (a + b - 1) / b; }

extern "C" void kernel_launch(void* const* d_in, const int* in_sizes, int n_in,
                              void* d_out, int out_size, void* d_ws, size_t ws_size,
                              hipStream_t stream) {
    const float* x        = (const float*)d_in[0];
    const int*   edge     = (const int*)  d_in[1];
    const float* bn_gamma = (const float*)d_in[2];
    const float* bn_beta  = (const float*)d_in[3];
    const float* bn_mean  = (const float*)d_in[4];
    const float* bn_var   = (const float*)d_in[5];
    const float* W_lin    = (const float*)d_in[6];
    const float* b_lin    = (const float*)d_in[7];
    const float* W1       = (const float*)d_in[8];
    const float* b1       = (const float*)d_in[9];
    const float* W2       = (const float*)d_in[10];
    const float* b2       = (const float*)d_in[11];
    const float* W_loc    = (const float*)d_in[12];
    const float* b_loc    = (const float*)d_in[13];
    const float* loc_vec  = (const float*)d_in[14];

    const int N = in_sizes[0] / N_IN;
    const int E = in_sizes[1] / 2;
    const int* src = edge;          // edge_index[0, :]
    const int* dst = edge + E;      // edge_index[1, :]

    // Workspace layout (floats)
    float* ws     = (float*)d_ws;
    float* bn_a   = ws;
    float* bn_c   = ws + 256;
    float* dinv_s = ws + 512;
    float* dinv_d = dinv_s + N;
    float* h      = dinv_d + N;                 // [N,128]
    float* thw    = h   + (size_t)N * N_HID;    // [N,128] h@W scratch
    float* acc    = thw + (size_t)N * N_HID;    // [N,128] aggregation scratch
    float* S      = acc + (size_t)N * N_HID;    // s1 then s2
    float* R      = S   + (size_t)N * N_HID;    // r1 then r2
    float* loc    = R   + (size_t)N * N_HID;    // [N,512]

    const int NH = N * N_HID;

    // --- BN fold + degrees ---
    bn_prep_kernel<<<1, 256, 0, stream>>>(bn_gamma, bn_beta, bn_mean, bn_var, bn_a, bn_c, N_IN);
    fill_kernel<<<cdiv(N, 256), 256, 0, stream>>>(dinv_s, 1.0f, N);
    fill_kernel<<<cdiv(N, 256), 256, 0, stream>>>(dinv_d, 1.0f, N);
    degree_kernel<<<cdiv(E, 256), 256, 0, stream>>>(src, dst, dinv_s, dinv_d, E);
    rsqrt_kernel<<<cdiv(N, 256), 256, 0, stream>>>(dinv_s, N);
    rsqrt_kernel<<<cdiv(N, 256), 256, 0, stream>>>(dinv_d, N);

    // --- h = BN(x) @ W_lin + b_lin ---
    {
        int tiles = cdiv(N, 16) * (N_HID / 16);
        gemm_wmma_f32<N_IN, N_HID, true><<<cdiv(tiles, 4), 128, 0, stream>>>(
            x, W_lin, b_lin, bn_a, bn_c, h, N);
    }

    const int gemmHH_blocks = cdiv(cdiv(N, 16) * (N_HID / 16), 4);
    const int edge_blocks   = cdiv(E, 8);   // 8 waves / 256-thread block
    const int nh_blocks     = cdiv(NH, 256);

    // --- supply direction: s1 = conv(h, W1); s2 = conv(s1, W2) ---
    gemm_wmma_f32<N_HID, N_HID, false><<<gemmHH_blocks, 128, 0, stream>>>(
        h, W1, nullptr, nullptr, nullptr, thw, N);
    fill_kernel<<<nh_blocks, 256, 0, stream>>>(acc, 0.0f, NH);
    scatter_kernel<<<edge_blocks, 256, 0, stream>>>(thw, acc, dinv_s, src, dst, E);
    finalize_kernel<<<nh_blocks, 256, 0, stream>>>(acc, thw, dinv_s, b1, S, NH);

    gemm_wmma_f32<N_HID, N_HID, false><<<gemmHH_blocks, 128, 0, stream>>>(
        S, W2, nullptr, nullptr, nullptr, thw, N);
    fill_kernel<<<nh_blocks, 256, 0, stream>>>(acc, 0.0f, NH);
    scatter_kernel<<<edge_blocks, 256, 0, stream>>>(thw, acc, dinv_s, src, dst, E);
    finalize_kernel<<<nh_blocks, 256, 0, stream>>>(acc, thw, dinv_s, b2, S, NH);

    // --- demand direction (reversed edges): r1 = conv(h, W1); r2 = conv(r1, W2) ---
    gemm_wmma_f32<N_HID, N_HID, false><<<gemmHH_blocks, 128, 0, stream>>>(
        h, W1, nullptr, nullptr, nullptr, thw, N);
    fill_kernel<<<nh_blocks, 256, 0, stream>>>(acc, 0.0f, NH);
    scatter_kernel<<<edge_blocks, 256, 0, stream>>>(thw, acc, dinv_d, dst, src, E);
    finalize_kernel<<<nh_blocks, 256, 0, stream>>>(acc, thw, dinv_d, b1, R, NH);

    gemm_wmma_f32<N_HID, N_HID, false><<<gemmHH_blocks, 128, 0, stream>>>(
        R, W2, nullptr, nullptr, nullptr, thw, N);
    fill_kernel<<<nh_blocks, 256, 0, stream>>>(acc, 0.0f, NH);
    scatter_kernel<<<edge_blocks, 256, 0, stream>>>(thw, acc, dinv_d, dst, src, E);
    finalize_kernel<<<nh_blocks, 256, 0, stream>>>(acc, thw, dinv_d, b2, R, NH);

    // --- loc = loc_vec @ W_loc + b_loc ---
    {
        int tiles = cdiv(N, 16) * (N_LOC / 16);
        gemm_wmma_f32<N_IN, N_LOC, false><<<cdiv(tiles, 4), 128, 0, stream>>>(
            loc_vec, W_loc, b_loc, nullptr, nullptr, loc, N);
    }

    // --- per-edge logits / cosine ---
    edge_kernel<<<edge_blocks, 256, 0, stream>>>(S, R, loc, src, dst, (float*)d_out, E);
}